// E3nnMLPBlockGate_63024350102048
// MI455X (gfx1250) — compile-verified
//
#include <hip/hip_runtime.h>
#include <math.h>

// ---------------- problem constants ----------------
#define MUL_S_IN 64
#define MUL_V_IN 32
#define MUL_S    64
#define MUL_G    32
#define MUL_V    32
#define NS       96          // MUL_S + MUL_G
#define XCOLS    160         // MUL_S_IN + 3*MUL_V_IN
#define NSTATS   224         // 96 sums + 96 sumsq + 32 v-sumsq
#define NB1      512         // fixed grid for the stats pass (partials layout)
#define NB3      1024        // fixed grid for the apply pass
#define EPS_F    1e-5f
#define INV_SQRT_S 0.125f                 // 1/sqrt(64)
#define INV_SQRT_V 0.17677669529663687f   // 1/sqrt(32)

// swizzled-LDS row strides (pad by 32 floats = 32 banks so the h=1 lane half
// is exactly 64-bank-disjoint from the h=0 half -> conflict-free ds_load_b64)
#define W0_ROW   224         // 96*2 + 32
#define W1_ROW   96          // 32*2 + 32

typedef float v8f __attribute__((ext_vector_type(8)));
typedef float v2f __attribute__((ext_vector_type(2)));

struct __attribute__((aligned(4))) f3 { float x, y, z; };

// fp32 WMMA: D(16x16) = A(16x4) * B(4x16) + C  -> v_wmma_f32_16x16x4_f32
__device__ __forceinline__ v8f wmma4f32(v2f a, v2f b, v8f c) {
  return __builtin_amdgcn_wmma_f32_16x16x4_f32(false, a, false, b,
                                               (short)0, c, false, false);
}

__device__ __forceinline__ float sigmoidf_fast(float x) {
  return 1.0f / (1.0f + __expf(-x));
}

// async global->LDS copy of one 16B chunk per lane (gfx1250 ASYNCcnt path)
__device__ __forceinline__ void async_copy_b128(const float* gsrc, float* ldst) {
  unsigned lds = (unsigned)(unsigned long long)(const void*)ldst;
  asm volatile("global_load_async_to_lds_b128 %0, %1, off"
               :: "v"(lds), "v"(gsrc) : "memory");
}
__device__ __forceinline__ void async_wait0() {
  asm volatile("s_wait_asynccnt 0x0" ::: "memory");
}

// stage W0/W1 through LDS scratch (async), then swizzle into fragment layout:
//   w0swz[p*W0_ROW + 2*c + r] = W0[(2p+r)*96 + c]   (p = K-pair index)
//   w1swz[p*W1_ROW + 2*c + r] = W1[(2p+r)*32 + c]
__device__ __forceinline__ void stage_weights(const float* __restrict__ W0,
                                              const float* __restrict__ W1,
                                              float* w0swz, float* w1swz,
                                              float* scratch, int tid) {
  // ---- W0: 64x96 = 6144 dwords = 256 threads * 6 * b128 ----
#pragma unroll
  for (int i = 0; i < 6; ++i) {
    int d = (i * 256 + tid) * 4;
    async_copy_b128(W0 + d, scratch + d);
  }
  async_wait0();
  __syncthreads();
  for (int idx = tid; idx < MUL_S_IN * NS; idx += 256) {
    int row = idx / NS, c = idx - row * NS;
    w0swz[(row >> 1) * W0_ROW + 2 * c + (row & 1)] = scratch[idx];
  }
  __syncthreads();
  // ---- W1: 32x32 = 1024 dwords = 256 threads * 1 * b128 ----
  async_copy_b128(W1 + tid * 4, scratch + tid * 4);
  async_wait0();
  __syncthreads();
  for (int idx = tid; idx < MUL_V_IN * MUL_V; idx += 256) {
    int row = idx >> 5, c = idx & 31;
    w1swz[(row >> 1) * W1_ROW + 2 * c + (row & 1)] = scratch[idx];
  }
  __syncthreads();
}

// =====================================================================
// Kernel 1: WMMA GEMMs, accumulate per-block statistics (deterministic)
// =====================================================================
__global__ __launch_bounds__(256) void k1_stats(const float* __restrict__ x,
                                                const float* __restrict__ W0,
                                                const float* __restrict__ W1,
                                                float* __restrict__ partials,
                                                int nrows) {
  __shared__ float w0swz[32 * W0_ROW];   // 28 KB
  __shared__ float w1swz[16 * W1_ROW];   // 6 KB
  __shared__ float scratch[6144];        // 24 KB: staging, then reduction

  const int tid = threadIdx.x;
  stage_weights(W0, W1, w0swz, w1swz, scratch, tid);

  const int lane = tid & 31;
  const int wv   = tid >> 5;
  const int l16  = lane & 15;
  const int h    = lane >> 4;

  const int w0base = h * W0_ROW + l16 * 2;   // + k*2*W0_ROW + t*32
  const int w1base = h * W1_ROW + l16 * 2;   // + k*2*W1_ROW + t*32

  float ssum[6] = {0.f,0.f,0.f,0.f,0.f,0.f};
  float ssq [6] = {0.f,0.f,0.f,0.f,0.f,0.f};
  float vsq [2] = {0.f,0.f};

  const int ntiles = nrows >> 4;   // 16 rows per tile
  for (int tile = blockIdx.x * 8 + wv; tile < ntiles; tile += NB1 * 8) {
    const float* xrow = x + (size_t)(tile * 16 + l16) * XCOLS;
    if (tile + NB1 * 8 < ntiles)   // wave-uniform; hint next tile into L2
      __builtin_prefetch(xrow + (size_t)(NB1 * 8) * 16 * XCOLS, 0, 0);

    // ---- GEMM1: s = s_in @ W0 ----
    // batch all 16 A-fragment loads first -> deep LOADcnt pipelining
    v2f a[16];
#pragma unroll
    for (int k = 0; k < 16; ++k)
      a[k] = *(const v2f*)(xrow + 4 * k + 2 * h);  // 8B aligned

    v8f acc[6] = {};
#pragma unroll
    for (int k = 0; k < 16; ++k) {
      const float* bp = w0swz + (k * 2 * W0_ROW + w0base);
#pragma unroll
      for (int t = 0; t < 6; ++t)
        acc[t] = wmma4f32(a[k], *(const v2f*)(bp + t * 32), acc[t]);
    }
#pragma unroll
    for (int t = 0; t < 6; ++t)
#pragma unroll
      for (int r = 0; r < 8; ++r) {
        float s = acc[t][r] * INV_SQRT_S;
        ssum[t] += s;
        ssq[t]  += s * s;
      }

    // ---- GEMM2: v[:,o,m] = v_in[:,:,m] @ W1, all 3 components ----
    // batch the 8 k-steps' 6-float A-slabs (compiler merges into b64 loads)
    v2f e01[8], e23[8], e45[8];
#pragma unroll
    for (int k = 0; k < 8; ++k) {
      const float* ap = xrow + MUL_S_IN + 3 * (4 * k + 2 * h);
      e01[k] = *(const v2f*)(ap + 0);
      e23[k] = *(const v2f*)(ap + 2);
      e45[k] = *(const v2f*)(ap + 4);
    }

    v8f vac[3][2] = {};
#pragma unroll
    for (int k = 0; k < 8; ++k) {
      v2f am0, am1, am2;
      am0.x = e01[k].x; am0.y = e23[k].y;
      am1.x = e01[k].y; am1.y = e45[k].x;
      am2.x = e23[k].x; am2.y = e45[k].y;
      const float* bp = w1swz + (k * 2 * W1_ROW + w1base);
#pragma unroll
      for (int t = 0; t < 2; ++t) {
        v2f b = *(const v2f*)(bp + t * 32);
        vac[0][t] = wmma4f32(am0, b, vac[0][t]);
        vac[1][t] = wmma4f32(am1, b, vac[1][t]);
        vac[2][t] = wmma4f32(am2, b, vac[2][t]);
      }
    }
#pragma unroll
    for (int m = 0; m < 3; ++m)
#pragma unroll
      for (int t = 0; t < 2; ++t)
#pragma unroll
        for (int r = 0; r < 8; ++r) {
          float v = vac[m][t][r] * INV_SQRT_V;
          vsq[t] += v * v;
        }
  }

  // ---- fixed-order block reduction (reuse scratch as red[2][8][NSTATS]) ----
  float* red = scratch;
  const int slotbase = (h * 8 + wv) * NSTATS;
#pragma unroll
  for (int t = 0; t < 6; ++t) {
    red[slotbase + t * 16 + l16]      = ssum[t];
    red[slotbase + 96 + t * 16 + l16] = ssq[t];
  }
#pragma unroll
  for (int t = 0; t < 2; ++t)
    red[slotbase + 192 + t * 16 + l16] = vsq[t];
  __syncthreads();

  if (tid < NSTATS) {
    float a = 0.f;
    for (int w = 0; w < 16; ++w)
      a += red[w * NSTATS + tid];
    partials[(size_t)blockIdx.x * NSTATS + tid] = a;
  }
}

// =====================================================================
// Kernel 2: finalize scale/shift from partials (single block)
// =====================================================================
__global__ __launch_bounds__(128) void k2_finalize(const float* __restrict__ partials,
                                                   const float* __restrict__ bnw,
                                                   const float* __restrict__ bnb,
                                                   float* __restrict__ stats,
                                                   int nrows) {
  const int t = threadIdx.x;
  const float invN = 1.0f / (float)nrows;
  if (t < NS) {
    float s1 = 0.f, s2 = 0.f;
    for (int b = 0; b < NB1; ++b) {
      s1 += partials[(size_t)b * NSTATS + t];
      s2 += partials[(size_t)b * NSTATS + 96 + t];
    }
    float mean  = s1 * invN;
    float var   = s2 * invN - mean * mean;
    float scale = rsqrtf(var + EPS_F) * bnw[t];
    stats[t]      = scale;
    stats[96 + t] = bnb[t] - mean * scale;
  } else if (t < NS + MUL_V) {
    int o = t - NS;
    float s2 = 0.f;
    for (int b = 0; b < NB1; ++b)
      s2 += partials[(size_t)b * NSTATS + 192 + o];
    float vn = s2 * invN * (1.0f / 3.0f);
    stats[192 + o] = rsqrtf(vn + EPS_F) * bnw[NS + o];
  }
}

// =====================================================================
// Kernel 3: recompute GEMMs, apply BN + SiLU/sigmoid gating, write out
// =====================================================================
__global__ __launch_bounds__(256) void k3_apply(const float* __restrict__ x,
                                                const float* __restrict__ W0,
                                                const float* __restrict__ W1,
                                                const float* __restrict__ stats,
                                                float* __restrict__ out,
                                                int nrows, float c_silu, float c_sig) {
  __shared__ float w0swz[32 * W0_ROW];
  __shared__ float w1swz[16 * W1_ROW];
  __shared__ float scratch[6144];

  const int tid = threadIdx.x;
  stage_weights(W0, W1, w0swz, w1swz, scratch, tid);

  const int lane = tid & 31;
  const int wv   = tid >> 5;
  const int l16  = lane & 15;
  const int h    = lane >> 4;

  const int w0base = h * W0_ROW + l16 * 2;
  const int w1base = h * W1_ROW + l16 * 2;

  // per-lane BN parameters for the 6 s-columns / 2 v-columns this lane owns
  float sscale[6], sshift[6], vscl[2];
#pragma unroll
  for (int t = 0; t < 6; ++t) {
    sscale[t] = stats[t * 16 + l16];
    sshift[t] = stats[96 + t * 16 + l16];
  }
#pragma unroll
  for (int t = 0; t < 2; ++t)
    vscl[t] = stats[192 + t * 16 + l16];

  const int ntiles = nrows >> 4;
  for (int tile = blockIdx.x * 8 + wv; tile < ntiles; tile += NB3 * 8) {
    const int rowBase = tile * 16;
    const float* xrow = x + (size_t)(rowBase + l16) * XCOLS;
    if (tile + NB3 * 8 < ntiles)   // wave-uniform
      __builtin_prefetch(xrow + (size_t)(NB3 * 8) * 16 * XCOLS, 0, 0);

    // ---- GEMM1 (batched A loads) ----
    v2f a[16];
#pragma unroll
    for (int k = 0; k < 16; ++k)
      a[k] = *(const v2f*)(xrow + 4 * k + 2 * h);

    v8f acc[6] = {};
#pragma unroll
    for (int k = 0; k < 16; ++k) {
      const float* bp = w0swz + (k * 2 * W0_ROW + w0base);
#pragma unroll
      for (int t = 0; t < 6; ++t)
        acc[t] = wmma4f32(a[k], *(const v2f*)(bp + t * 32), acc[t]);
    }

    // ---- BN + activations; scalars stored, gates kept in registers ----
    float gate[2][8];
#pragma unroll
    for (int t = 0; t < 6; ++t) {
#pragma unroll
      for (int r = 0; r < 8; ++r) {
        float sbn = acc[t][r] * INV_SQRT_S * sscale[t] + sshift[t];
        if (t < 4) {
          int row = rowBase + 8 * h + r;
          out[(size_t)row * XCOLS + t * 16 + l16] =
              sbn * sigmoidf_fast(sbn) * c_silu;
        } else {
          gate[t - 4][r] = sigmoidf_fast(sbn) * c_sig;
        }
      }
    }

    // ---- GEMM2 (all 3 components, batched A loads) + gating ----
    v2f e01[8], e23[8], e45[8];
#pragma unroll
    for (int k = 0; k < 8; ++k) {
      const float* ap = xrow + MUL_S_IN + 3 * (4 * k + 2 * h);
      e01[k] = *(const v2f*)(ap + 0);
      e23[k] = *(const v2f*)(ap + 2);
      e45[k] = *(const v2f*)(ap + 4);
    }

    v8f vac[3][2] = {};
#pragma unroll
    for (int k = 0; k < 8; ++k) {
      v2f am0, am1, am2;
      am0.x = e01[k].x; am0.y = e23[k].y;
      am1.x = e01[k].y; am1.y = e45[k].x;
      am2.x = e23[k].x; am2.y = e45[k].y;
      const float* bp = w1swz + (k * 2 * W1_ROW + w1base);
#pragma unroll
      for (int t = 0; t < 2; ++t) {
        v2f b = *(const v2f*)(bp + t * 32);
        vac[0][t] = wmma4f32(am0, b, vac[0][t]);
        vac[1][t] = wmma4f32(am1, b, vac[1][t]);
        vac[2][t] = wmma4f32(am2, b, vac[2][t]);
      }
    }
#pragma unroll
    for (int t = 0; t < 2; ++t) {
      const int o = t * 16 + l16;
      const float vs = vscl[t] * INV_SQRT_V;
#pragma unroll
      for (int r = 0; r < 8; ++r) {
        int row = rowBase + 8 * h + r;
        float f = vs * gate[t][r];
        f3 g;
        g.x = vac[0][t][r] * f;
        g.y = vac[1][t][r] * f;
        g.z = vac[2][t][r] * f;
        *(f3*)(out + (size_t)row * XCOLS + MUL_S_IN + 3 * o) = g;
      }
    }
  }
}

// ---------------- host: replicate reference's trapz constants ----------------
static void norm2moms(float* c_silu, float* c_sig) {
  const int M = 200001;
  const double dz = 20.0 / (double)(M - 1);
  double m2s = 0.0, m2g = 0.0, prev_s = 0.0, prev_g = 0.0;
  for (int i = 0; i < M; ++i) {
    double z   = -10.0 + dz * (double)i;
    double pdf = exp(-0.5 * z * z) / sqrt(2.0 * M_PI);
    double sg  = 1.0 / (1.0 + exp(-z));
    double sl  = z * sg;
    double ys = sl * sl * pdf, yg = sg * sg * pdf;
    if (i) { m2s += 0.5 * (ys + prev_s) * dz; m2g += 0.5 * (yg + prev_g) * dz; }
    prev_s = ys; prev_g = yg;
  }
  *c_silu = (float)(1.0 / sqrt(m2s));
  *c_sig  = (float)(1.0 / sqrt(m2g));
}

extern "C" void kernel_launch(void* const* d_in, const int* in_sizes, int n_in,
                              void* d_out, int out_size, void* d_ws, size_t ws_size,
                              hipStream_t stream) {
  const float* x   = (const float*)d_in[0];
  const float* W0  = (const float*)d_in[1];
  const float* W1  = (const float*)d_in[2];
  const float* bnw = (const float*)d_in[3];
  const float* bnb = (const float*)d_in[4];
  float* out = (float*)d_out;

  const int nrows = in_sizes[0] / XCOLS;   // 524288 (multiple of 16)

  float* partials = (float*)d_ws;                    // NB1 * NSTATS floats
  float* stats    = partials + (size_t)NB1 * NSTATS; // NSTATS floats

  float c_silu, c_sig;
  norm2moms(&c_silu, &c_sig);

  k1_stats<<<NB1, 256, 0, stream>>>(x, W0, W1, partials, nrows);
  k2_finalize<<<1, 128, 0, stream>>>(partials, bnw, bnb, stats, nrows);
  k3_apply<<<NB3, 256, 0, stream>>>(x, W0, W1, stats, out, nrows, c_silu, c_sig);
}